// MaskAbsorbingModel_64029372448802
// MI455X (gfx1250) — compile-verified
//
#include <hip/hip_runtime.h>
#include <hip/hip_bf16.h>

// ---------------------------------------------------------------------------
// Constants matching the reference
// ---------------------------------------------------------------------------
#define N_ATOMS     5
#define T_TRIP      8
#define NUM_EXPERTS 18
#define MAX_BEADS   4
#define X_COLS      38
#define ROW_F       15              // N_ATOMS*3 floats per row
#define EPS_F       1e-8f
#define ML2_F       (0.02f * 0.02f) // MIN_LEN^2

#define TILE_ROWS   256
#define BLOCK_THR   256
#define NUM_BLOCKS  1024            // must be a multiple of 64 for the WMMA reduce

typedef __attribute__((ext_vector_type(2))) float v2f;
typedef __attribute__((ext_vector_type(8))) float v8f;

// ---------------------------------------------------------------------------
// Main streaming kernel: each block stages a 256-row tile of true/pred/mask
// into LDS with coalesced b128 loads, then each thread evaluates one row's
// atom-MSE + angle loss. Per-block partial sums go to d_ws (deterministic).
// ---------------------------------------------------------------------------
__global__ __launch_bounds__(BLOCK_THR)
void loss_main(const float* __restrict__ x,
               const float* __restrict__ yt,
               const float* __restrict__ yp,
               const float* __restrict__ mk,
               const int*   __restrict__ trip,
               const int*   __restrict__ vld,
               float*       __restrict__ partials,
               int N) {
    __shared__ __align__(16) float s_t[TILE_ROWS * ROW_F];
    __shared__ __align__(16) float s_p[TILE_ROWS * ROW_F];
    __shared__ __align__(16) float s_m[TILE_ROWS * ROW_F];
    __shared__ int   s_trip[NUM_EXPERTS * MAX_BEADS * T_TRIP * 3]; // 1728
    __shared__ int   s_vld [NUM_EXPERTS * MAX_BEADS * T_TRIP];     // 576
    __shared__ float s_red[BLOCK_THR];

    const int tid = threadIdx.x;

    // Stage lookup tables once per block.
    for (int i = tid; i < NUM_EXPERTS * MAX_BEADS * T_TRIP * 3; i += BLOCK_THR)
        s_trip[i] = trip[i];
    for (int i = tid; i < NUM_EXPERTS * MAX_BEADS * T_TRIP; i += BLOCK_THR)
        s_vld[i] = vld[i];

    const int numTiles = (N + TILE_ROWS - 1) / TILE_ROWS;
    float acc = 0.0f;

    for (int tile = blockIdx.x; tile < numTiles; tile += gridDim.x) {
        const int tileStart = tile * TILE_ROWS;
        const int rowsInTile = (N - tileStart) < TILE_ROWS ? (N - tileStart) : TILE_ROWS;
        const int nElem = rowsInTile * ROW_F;
        const int nV4 = nElem >> 2;

        // Prefetch next tile (gfx1250 global_prefetch_b8).
        {
            int nextTile = tile + gridDim.x;
            if (nextTile < numTiles) {
                size_t nb = (size_t)nextTile * TILE_ROWS * ROW_F;
                __builtin_prefetch(yt + nb, 0, 0);
                __builtin_prefetch(yp + nb, 0, 0);
                __builtin_prefetch(mk + nb, 0, 0);
            }
        }

        // Coalesced b128 staging: tile base byte offset = tileStart*60, and
        // tileStart is a multiple of 256 -> 16B aligned.
        {
            const size_t gBase4 = ((size_t)tileStart * ROW_F) >> 2;
            const float4* gt4 = reinterpret_cast<const float4*>(yt) + gBase4;
            const float4* gp4 = reinterpret_cast<const float4*>(yp) + gBase4;
            const float4* gm4 = reinterpret_cast<const float4*>(mk) + gBase4;
            float4* st4 = reinterpret_cast<float4*>(s_t);
            float4* sp4 = reinterpret_cast<float4*>(s_p);
            float4* sm4 = reinterpret_cast<float4*>(s_m);
            for (int i = tid; i < nV4; i += BLOCK_THR) {
                st4[i] = gt4[i];
                sp4[i] = gp4[i];
                sm4[i] = gm4[i];
            }
            // scalar tail (only possible if rowsInTile % 4 != 0)
            for (int i = (nV4 << 2) + tid; i < nElem; i += BLOCK_THR) {
                size_t g = (size_t)tileStart * ROW_F + i;
                s_t[i] = yt[g];
                s_p[i] = yp[g];
                s_m[i] = mk[g];
            }
        }
        __syncthreads();

        if (tid < rowsInTile) {
            const int rb = tid * ROW_F;
            const size_t row = (size_t)tileStart + tid;

            // Only the two gating columns of x (8B aligned -> one b64 load).
            float2 bg = *reinterpret_cast<const float2*>(x + row * X_COLS + 36);
            float bead = bg.x;
            float gate = bg.y;

            // ---------------- atom MSE ----------------
            float num = 0.0f, den = 0.0f;
            int avmask = 0;
#pragma unroll
            for (int a = 0; a < N_ATOMS; ++a) {
                const int o = rb + 3 * a;
                float m0 = s_m[o], m1 = s_m[o + 1], m2 = s_m[o + 2];
                float d0 = s_t[o]     - s_p[o];
                float d1 = s_t[o + 1] - s_p[o + 1];
                float d2 = s_t[o + 2] - s_p[o + 2];
                float se = d0 * d0 * m0 + d1 * d1 * m1 + d2 * d2 * m2;
                bool v = (m0 > 0.0f) | (m1 > 0.0f) | (m2 > 0.0f);
                if (v) { num += se; den += 1.0f; avmask |= (1 << a); }
            }
            const float rowAtom = num / (den + EPS_F);

            // ---------------- angle loss ----------------
            float g = __builtin_isfinite(gate) ? gate : 0.0f;
            float b = __builtin_isfinite(bead) ? bead : 0.0f;
            float rg = rintf(g) - 1.0f;
            rg = fminf(fmaxf(rg, 0.0f), (float)(NUM_EXPERTS - 1));
            float rbd = rintf(b);
            rbd = fminf(fmaxf(rbd, 0.0f), (float)(MAX_BEADS - 1));
            const int res_id  = (int)rg;
            const int bead_id = (int)rbd;
            const int vb = (res_id * MAX_BEADS + bead_id) * T_TRIP;
            const int tb = vb * 3;

            float anum = 0.0f, aden = 0.0f;
#pragma unroll
            for (int t = 0; t < T_TRIP; ++t) {
                const int i0 = s_trip[tb + 3 * t];
                const int i1 = s_trip[tb + 3 * t + 1];
                const int i2 = s_trip[tb + 3 * t + 2];
                const bool tok = (i0 >= 0) && (i1 >= 0) && (i2 >= 0);
                const int bv = s_vld[vb + t];
                const int ii = i0 < 0 ? 0 : i0;
                const int jj = i1 < 0 ? 0 : i1;
                const int kk = i2 < 0 ? 0 : i2;
                const bool vi = (avmask >> ii) & 1;
                const bool vj = (avmask >> jj) & 1;
                const bool vk = (avmask >> kk) & 1;

                const int oi = rb + 3 * ii, oj = rb + 3 * jj, oq = rb + 3 * kk;

                float v1tx = s_t[oi]     - s_t[oj];
                float v1ty = s_t[oi + 1] - s_t[oj + 1];
                float v1tz = s_t[oi + 2] - s_t[oj + 2];
                float v2tx = s_t[oq]     - s_t[oj];
                float v2ty = s_t[oq + 1] - s_t[oj + 1];
                float v2tz = s_t[oq + 2] - s_t[oj + 2];
                float v1px = s_p[oi]     - s_p[oj];
                float v1py = s_p[oi + 1] - s_p[oj + 1];
                float v1pz = s_p[oi + 2] - s_p[oj + 2];
                float v2px = s_p[oq]     - s_p[oj];
                float v2py = s_p[oq + 1] - s_p[oj + 1];
                float v2pz = s_p[oq + 2] - s_p[oj + 2];

                float l1t = v1tx * v1tx + v1ty * v1ty + v1tz * v1tz;
                float l2t = v2tx * v2tx + v2ty * v2ty + v2tz * v2tz;
                float l1p = v1px * v1px + v1py * v1py + v1pz * v1pz;
                float l2p = v2px * v2px + v2py * v2py + v2pz * v2pz;

                const bool lenok = (l1t > ML2_F) && (l2t > ML2_F) &&
                                   (l1p > ML2_F) && (l2p > ML2_F);
                const bool valid = tok && (bv > 0) && vi && vj && vk && lenok;

                float r1t = 1.0f / sqrtf(fmaxf(l1t, ML2_F));
                float r2t = 1.0f / sqrtf(fmaxf(l2t, ML2_F));
                float r1p = 1.0f / sqrtf(fmaxf(l1p, ML2_F));
                float r2p = 1.0f / sqrtf(fmaxf(l2p, ML2_F));

                float u1tx = v1tx * r1t, u1ty = v1ty * r1t, u1tz = v1tz * r1t;
                float u2tx = v2tx * r2t, u2ty = v2ty * r2t, u2tz = v2tz * r2t;
                float u1px = v1px * r1p, u1py = v1py * r1p, u1pz = v1pz * r1p;
                float u2px = v2px * r2p, u2py = v2py * r2p, u2pz = v2pz * r2p;

                float ct = u1tx * u2tx + u1ty * u2ty + u1tz * u2tz;
                float cp = u1px * u2px + u1py * u2py + u1pz * u2pz;
                ct = fminf(fmaxf(ct, -1.0f + 1e-6f), 1.0f - 1e-6f);
                cp = fminf(fmaxf(cp, -1.0f + 1e-6f), 1.0f - 1e-6f);

                float ctx = u1ty * u2tz - u1tz * u2ty;
                float cty = u1tz * u2tx - u1tx * u2tz;
                float ctz = u1tx * u2ty - u1ty * u2tx;
                float st = sqrtf(ctx * ctx + cty * cty + ctz * ctz);

                float cpx = u1py * u2pz - u1pz * u2py;
                float cpy = u1pz * u2px - u1px * u2pz;
                float cpz = u1px * u2py - u1py * u2px;
                float sp = sqrtf(cpx * cpx + cpy * cpy + cpz * cpz);

                float dc = cp - ct, ds = sp - st;
                float per = dc * dc + ds * ds;
                if (valid) { anum += per; aden += 1.0f; }
            }
            acc += rowAtom + anum / (aden + EPS_F);
        }
        __syncthreads();
    }

    // Deterministic block reduction.
    s_red[tid] = acc;
    __syncthreads();
#pragma unroll
    for (int s = BLOCK_THR / 2; s > 0; s >>= 1) {
        if (tid < s) s_red[tid] += s_red[tid + s];
        __syncthreads();
    }
    if (tid == 0) partials[blockIdx.x] = s_red[0];
}

// ---------------------------------------------------------------------------
// Final reduction on the matrix pipe: V_WMMA_F32_16X16X4_F32 with an all-ones
// B matrix folds 64 partials per issue into the f32 accumulator (exact f32
// FMAs, deterministic). One wave32, EXEC all-ones as WMMA requires.
// ---------------------------------------------------------------------------
__global__ __launch_bounds__(32)
void final_reduce_wmma(const float* __restrict__ partials,
                       float* __restrict__ out, float invN) {
    const int lane = threadIdx.x;
    v8f c = {};
    v2f bones;
    bones[0] = 1.0f;
    bones[1] = 1.0f;
    const float2* p2 = reinterpret_cast<const float2*>(partials);
#pragma unroll
    for (int base = 0; base < NUM_BLOCKS; base += 64) {
        float2 t = p2[(base >> 1) + lane];
        v2f a;
        a[0] = t.x;
        a[1] = t.y;
        // D = A(16x4 f32) x B(4x16 ones) + C  ->  every D column holds A row-sums
        c = __builtin_amdgcn_wmma_f32_16x16x4_f32(
            /*neg_a=*/false, a, /*neg_b=*/false, bones,
            /*c_mod=*/(short)0, c, /*reuse_a=*/false, /*reuse_b=*/false);
    }
    // Column 0 of D: M=0..7 on lane 0 (vgpr 0..7), M=8..15 on lane 16.
    float s = c[0] + c[1] + c[2] + c[3] + c[4] + c[5] + c[6] + c[7];
    float tot = __shfl(s, 0, 32) + __shfl(s, 16, 32);
    if (lane == 0) out[0] = tot * invN;
}

// ---------------------------------------------------------------------------
extern "C" void kernel_launch(void* const* d_in, const int* in_sizes, int n_in,
                              void* d_out, int out_size, void* d_ws, size_t ws_size,
                              hipStream_t stream) {
    const float* x    = (const float*)d_in[0];
    const float* yt   = (const float*)d_in[1];
    const float* yp   = (const float*)d_in[2];
    const float* mk   = (const float*)d_in[3];
    const int*   trip = (const int*)d_in[4];
    const int*   vld  = (const int*)d_in[5];
    float* out      = (float*)d_out;
    float* partials = (float*)d_ws;

    const int N = in_sizes[1] / ROW_F;

    loss_main<<<NUM_BLOCKS, BLOCK_THR, 0, stream>>>(x, yt, yp, mk, trip, vld,
                                                    partials, N);
    final_reduce_wmma<<<1, 32, 0, stream>>>(partials, out, 1.0f / (float)N);
}